// TransMILSquaring_40561671144063
// MI455X (gfx1250) — compile-verified
//
#include <hip/hip_runtime.h>
#include <hip/hip_bf16.h>

// ---------------------------------------------------------------------------
// TransMIL forward on MI455X (gfx1250).  Big GEMMs: V_WMMA_F32_16X16X4_F32
// (fp32 matrix cores, wave32) with LDS-staged tiles filled by the CDNA5 async
// copy unit (GLOBAL_LOAD_ASYNC_TO_LDS_B128, ASYNCcnt / s_wait_asynccnt).
// Inner loop: ds_load + v_wmma only.
// ---------------------------------------------------------------------------

typedef __attribute__((ext_vector_type(2))) float v2f;
typedef __attribute__((ext_vector_type(8))) float v8f;

// ---- problem constants ----------------------------------------------------
constexpr long B_    = 2;
constexpr long NTOK  = 10000;
constexpr long DIN   = 768;
constexpr long C_    = 512;
constexpr long H_    = 8;
constexpr long DH_   = 64;
constexpr long MLM   = 256;          // landmarks (C/2)
constexpr long SEQ1  = NTOK + 1;     // 10001 (cls + tokens; M = N*N - n = 0)
constexpr long PAD   = 239;          // (256 - 10001%256)%256
constexpr long NPD   = 10240;        // padded length
constexpr long LSEG  = NPD / MLM;    // 40 positions per landmark

// ---- workspace layout (float offsets) -------------------------------------
constexpr size_t SZ_X1  = (size_t)B_*SEQ1*C_;
constexpr size_t SZ_XP  = (size_t)B_*NPD*C_;
constexpr size_t SZ_QKV = (size_t)B_*NPD*3*C_;
constexpr size_t SZ_SIM = (size_t)16*NPD*MLM;        // reused: sim3, sim1, proj, ppeg tmp
constexpr size_t SZ_LM  = (size_t)16*2*MLM*DH_;
constexpr size_t SZ_MM  = (size_t)16*MLM*MLM;
constexpr size_t SZ_T   = (size_t)16*MLM*DH_;

constexpr size_t OFF_X1  = 0;
constexpr size_t OFF_XP  = OFF_X1 + SZ_X1;
constexpr size_t OFF_QKV = OFF_XP + SZ_XP;
constexpr size_t OFF_SIM = OFF_QKV + SZ_QKV;
constexpr size_t OFF_LM  = OFF_SIM + SZ_SIM;
constexpr size_t OFF_A2  = OFF_LM + SZ_LM;
constexpr size_t OFF_Z   = OFF_A2 + SZ_MM;
constexpr size_t OFF_Z2  = OFF_Z + SZ_MM;
constexpr size_t OFF_G1  = OFF_Z2 + SZ_MM;
constexpr size_t OFF_P   = OFF_G1 + SZ_MM;
constexpr size_t OFF_E   = OFF_P + SZ_MM;
constexpr size_t OFF_T   = OFF_E + SZ_MM;
constexpr size_t OFF_U   = OFF_T + SZ_T;
constexpr size_t OFF_SC  = OFF_U + SZ_T;             // scalars

// ---------------------------------------------------------------------------
// WMMA helper: D = A(16x4) * B(4x16) + C, fp32.
// A frag (lane): m = lane&15, kbase = 2*(lane>>4): a.x=A[m][k+kb], a.y=A[m][k+kb+1]
// B frag (lane): n = lane&15, kbase = 2*(lane>>4): b.x=B[k+kb][n], b.y=B[k+kb+1][n]
// C/D:  vgpr r -> row = r + 8*(lane>>4), col = lane&15
// ---------------------------------------------------------------------------
__device__ __forceinline__ v8f wmma4(v2f a, v2f b, v8f c) {
  return __builtin_amdgcn_wmma_f32_16x16x4_f32(false, a, false, b, (short)0, c,
                                               false, false);
}

// CDNA5 async global -> LDS copy (16B), tracked with ASYNCcnt.
// lds_off is a byte offset into the (dynamic) LDS allocation.
__device__ __forceinline__ void async_b128(unsigned lds_off, const float* g) {
  asm volatile("global_load_async_to_lds_b128 %0, %1, off"
               :: "v"(lds_off), "v"(g) : "memory");
}
__device__ __forceinline__ void async_wait0() {
  asm volatile("s_wait_asynccnt 0x0" ::: "memory");
}

constexpr int KT  = 32;   // K stage width
constexpr int LDA = 36;   // A-tile LDS row stride (floats): 144B = 9*16B keeps
                          // b128 alignment, breaks ds bank conflicts
constexpr int SMEM_NN = (128 * LDA + KT * 16) * 4;   // A + k-major B tile
constexpr int SMEM_NT = (128 * LDA + 16 * LDA) * 4;  // A + n-major B tile

// flags: bit0 = ReLU, bit1 = fc1 row remap (20000 rows -> X1 rows w/ cls gap)
// Block: 256 threads = 8 waves; C-tile 128(M) x 16(N); K staged via LDS.
__global__ void gemm_nn(const float* __restrict__ A, long lda, long sAb, long sAh,
                        const float* __restrict__ B, long ldb, long sBb, long sBh,
                        float* __restrict__ C, long ldc, long sCb, long sCh,
                        int M, int N, int K, const float* __restrict__ bias,
                        float alpha, int flags, int zdiv) {
  extern __shared__ float smem[];
  float* As = smem;                 // [128][LDA]
  // B tile k-major: smem[128*LDA + k*16 + n], k in [0,32), n in [0,16)
  const int BOFF = 128 * LDA;

  int z  = blockIdx.z;
  int zb = z / zdiv, zh = z % zdiv;
  const float* Ab = A + (size_t)zb * sAb + (size_t)zh * sAh;
  const float* Bb = B + (size_t)zb * sBb + (size_t)zh * sBh;
  float*       Cb = C + (size_t)zb * sCb + (size_t)zh * sCh;

  int tid  = threadIdx.x;
  int lane = tid & 31, wave = tid >> 5;
  int tm = blockIdx.y * 8 + wave, tn = blockIdx.x;
  bool valid = (tm * 16 < M);
  int m0 = blockIdx.y * 128;
  int half = lane >> 4, l16 = lane & 15;

  v8f acc = {0.f, 0.f, 0.f, 0.f, 0.f, 0.f, 0.f, 0.f};

  for (int k0 = 0; k0 < K; k0 += KT) {
    // A tile 128x32: 1024 x b128, 4 per thread (coalesced along k)
#pragma unroll
    for (int i = 0; i < 4; ++i) {
      int idx = tid + i * 256;
      int r = idx >> 3, c4 = (idx & 7) * 4;
      int gr = m0 + r; if (gr >= M) gr = M - 1;
      async_b128((unsigned)((r * LDA + c4) * 4), Ab + (size_t)gr * lda + k0 + c4);
    }
    // B tile 32(k) x 16(n): 128 x b128, threads 0..127
    if (tid < 128) {
      int k = tid >> 2, c4 = (tid & 3) * 4;
      async_b128((unsigned)((BOFF + k * 16 + c4) * 4),
                 Bb + (size_t)(k0 + k) * ldb + tn * 16 + c4);
    }
    async_wait0();
    __syncthreads();
    int arow  = (wave * 16 + l16) * LDA + 2 * half;
    int bbase = BOFF + 32 * half + l16;      // B[k=kk+2*half][n=l16]
#pragma unroll
    for (int kk = 0; kk < KT; kk += 4) {
      v2f a; a.x = As[arow + kk];          a.y = As[arow + kk + 1];
      v2f b; b.x = smem[bbase + kk * 16];  b.y = smem[bbase + kk * 16 + 16];
      acc = wmma4(a, b, acc);
    }
    __syncthreads();
  }

  if (valid) {
    int ccol = tn * 16 + l16;
    float bv = bias ? bias[ccol] : 0.f;
#pragma unroll
    for (int r = 0; r < 8; ++r) {
      int crow = tm * 16 + r + 8 * half;
      float v = acc[r] * alpha + bv;
      if (flags & 1) v = fmaxf(v, 0.f);
      size_t orow = (size_t)crow;
      if (flags & 2) {  // fc1 remap: batch*10001 + 1 + row%10000
        size_t bt = (size_t)crow / 10000;
        orow = bt * 10001 + 1 + ((size_t)crow - bt * 10000);
      }
      Cb[orow * ldc + ccol] = v;
    }
  }
}

// C = alpha * A(MxK) * B^T, B stored N x K row-major.  Same tiling.
__global__ void gemm_nt(const float* __restrict__ A, long lda, long sAb, long sAh,
                        const float* __restrict__ B, long ldb, long sBb, long sBh,
                        float* __restrict__ C, long ldc, long sCb, long sCh,
                        int M, int N, int K, float alpha, int zdiv) {
  extern __shared__ float smem[];
  float* As = smem;                 // [128][LDA]
  const int BOFF = 128 * LDA;       // Bst[n][k], stride LDA

  int z  = blockIdx.z;
  int zb = z / zdiv, zh = z % zdiv;
  const float* Ab = A + (size_t)zb * sAb + (size_t)zh * sAh;
  const float* Bb = B + (size_t)zb * sBb + (size_t)zh * sBh;
  float*       Cb = C + (size_t)zb * sCb + (size_t)zh * sCh;

  int tid  = threadIdx.x;
  int lane = tid & 31, wave = tid >> 5;
  int tm = blockIdx.y * 8 + wave, tn = blockIdx.x;
  bool valid = (tm * 16 < M);
  int m0 = blockIdx.y * 128;
  int half = lane >> 4, l16 = lane & 15;

  v8f acc = {0.f, 0.f, 0.f, 0.f, 0.f, 0.f, 0.f, 0.f};

  for (int k0 = 0; k0 < K; k0 += KT) {
#pragma unroll
    for (int i = 0; i < 4; ++i) {
      int idx = tid + i * 256;
      int r = idx >> 3, c4 = (idx & 7) * 4;
      int gr = m0 + r; if (gr >= M) gr = M - 1;
      async_b128((unsigned)((r * LDA + c4) * 4), Ab + (size_t)gr * lda + k0 + c4);
    }
    // Bt tile 16(n) x 32(k): 128 x b128, threads 0..127 (coalesced along k)
    if (tid < 128) {
      int n = tid >> 3, c4 = (tid & 7) * 4;
      async_b128((unsigned)((BOFF + n * LDA + c4) * 4),
                 Bb + (size_t)(tn * 16 + n) * ldb + k0 + c4);
    }
    async_wait0();
    __syncthreads();
    int arow = (wave * 16 + l16) * LDA + 2 * half;
    int brow = BOFF + l16 * LDA + 2 * half;
#pragma unroll
    for (int kk = 0; kk < KT; kk += 4) {
      v2f a; a.x = As[arow + kk];   a.y = As[arow + kk + 1];
      v2f b; b.x = smem[brow + kk]; b.y = smem[brow + kk + 1];
      acc = wmma4(a, b, acc);
    }
    __syncthreads();
  }

  if (valid) {
    int ccol = tn * 16 + l16;
#pragma unroll
    for (int r = 0; r < 8; ++r) {
      int crow = tm * 16 + r + 8 * half;
      Cb[(size_t)crow * ldc + ccol] = acc[r] * alpha;
    }
  }
}

// ---------------------------------------------------------------------------
// elementwise / reduction kernels
// ---------------------------------------------------------------------------
__global__ void cls_init(float* __restrict__ X1, const float* __restrict__ cls) {
  int t = threadIdx.x + blockIdx.x * 256;
  if (t < 512) {
    X1[t] = cls[t];
    X1[(size_t)SEQ1 * C_ + t] = cls[t];
  }
}

__global__ void ln_pad_kernel(const float* __restrict__ X, const float* __restrict__ g,
                              const float* __restrict__ bb, float* __restrict__ XP) {
  int orow = blockIdx.x, b = blockIdx.y, t = threadIdx.x;
  float* out = XP + ((size_t)b * NPD + orow) * C_;
  if (orow < PAD) { out[t] = 0.f; out[t + 256] = 0.f; return; }
  const float* in = X + ((size_t)b * SEQ1 + (orow - PAD)) * C_;
  float x0 = in[t], x1 = in[t + 256];
  __shared__ float red[256];
  red[t] = x0 + x1; __syncthreads();
  for (int s = 128; s > 0; s >>= 1) { if (t < s) red[t] += red[t + s]; __syncthreads(); }
  float mu = red[0] / 512.f; __syncthreads();
  red[t] = (x0 - mu) * (x0 - mu) + (x1 - mu) * (x1 - mu); __syncthreads();
  for (int s = 128; s > 0; s >>= 1) { if (t < s) red[t] += red[t + s]; __syncthreads(); }
  float inv = rsqrtf(red[0] / 512.f + 1e-5f);
  out[t]       = (x0 - mu) * inv * g[t] + bb[t];
  out[t + 256] = (x1 - mu) * inv * g[t + 256] + bb[t + 256];
}

__global__ void landmark_kernel(const float* __restrict__ QKV, float* __restrict__ LM) {
  int i = blockIdx.x, z = blockIdx.y;          // landmark, bh
  int b = z >> 3, h = z & 7, d = threadIdx.x;  // d in [0,64)
  size_t base = ((size_t)b * NPD + (size_t)i * LSEG) * (3 * C_) + (size_t)h * DH_ + d;
  float sq = 0.f, sk = 0.f;
  for (int j = 0; j < LSEG; ++j) {
    sq += QKV[base + (size_t)j * (3 * C_)];
    sk += QKV[base + (size_t)j * (3 * C_) + C_];
  }
  size_t o = (size_t)z * 2 * MLM * DH_ + (size_t)i * DH_ + d;
  LM[o]             = sq * (0.125f / (float)LSEG);  // q landmarks pre-scaled by dh^-0.5
  LM[o + MLM * DH_] = sk * (1.f / (float)LSEG);
}

__global__ void softmax_rows(float* __restrict__ X, int ncols, long ld, long zstride) {
  float* row = X + (size_t)blockIdx.y * zstride + (size_t)blockIdx.x * ld;
  int t = threadIdx.x;
  __shared__ float red[256];
  float m = -3.402823e38f;
  for (int c = t; c < ncols; c += 256) m = fmaxf(m, row[c]);
  red[t] = m; __syncthreads();
  for (int s = 128; s > 0; s >>= 1) { if (t < s) red[t] = fmaxf(red[t], red[t + s]); __syncthreads(); }
  m = red[0]; __syncthreads();
  float sum = 0.f;
  for (int c = t; c < ncols; c += 256) { float e = expf(row[c] - m); row[c] = e; sum += e; }
  red[t] = sum; __syncthreads();
  for (int s = 128; s > 0; s >>= 1) { if (t < s) red[t] += red[t + s]; __syncthreads(); }
  float inv = 1.f / red[0];
  for (int c = t; c < ncols; c += 256) row[c] *= inv;
}

__global__ void init_scal(float* __restrict__ S) {
  if (threadIdx.x < 8) S[threadIdx.x] = 0.f;
}

// global (over all 16 bh) max row-sum and max col-sum of |A2|
__global__ void pinv_absmax(const float* __restrict__ A2, float* __restrict__ S) {
  int j = blockIdx.x, z = blockIdx.y, t = threadIdx.x;
  const float* M = A2 + (size_t)z * MLM * MLM;
  __shared__ float rc[256], rr[256];
  rc[t] = fabsf(M[(size_t)t * MLM + j]);   // column j
  rr[t] = fabsf(M[(size_t)j * MLM + t]);   // row j
  __syncthreads();
  for (int s = 128; s > 0; s >>= 1) {
    if (t < s) { rc[t] += rc[t + s]; rr[t] += rr[t + s]; }
    __syncthreads();
  }
  if (t == 0) {  // positive floats: int compare == float compare
    atomicMax((int*)&S[0], __float_as_int(rr[0]));  // max row-sum ("col" in ref)
    atomicMax((int*)&S[1], __float_as_int(rc[0]));  // max col-sum ("row" in ref)
  }
}

__global__ void pinv_make_z(const float* __restrict__ A2, const float* __restrict__ S,
                            float* __restrict__ Z) {
  int i = blockIdx.x, z = blockIdx.y, j = threadIdx.x;
  float rcp = 1.f / (S[0] * S[1]);
  Z[(size_t)z * MLM * MLM + (size_t)i * MLM + j] =
      A2[(size_t)z * MLM * MLM + (size_t)j * MLM + i] * rcp;
}

__global__ void diag_sub(const float* __restrict__ P, float* __restrict__ E, float alpha) {
  int i = blockIdx.x, z = blockIdx.y, j = threadIdx.x;
  size_t o = (size_t)z * MLM * MLM + (size_t)i * MLM + j;
  E[o] = (i == j ? alpha : 0.f) - P[o];
}

__global__ void res_conv_add(const float* __restrict__ QKV, const float* __restrict__ W,
                             float* __restrict__ OUT) {
  int p = blockIdx.x, z = blockIdx.y;
  int b = z >> 3, h = z & 7, d = threadIdx.x;
  float acc = 0.f;
  for (int k = 0; k < 33; ++k) {
    int pp = p + k - 16;
    if (pp >= 0 && pp < NPD)
      acc += W[h * 33 + k] * QKV[((size_t)b * NPD + pp) * (3 * C_) + 2 * C_ + h * DH_ + d];
  }
  OUT[((size_t)b * NPD + p) * C_ + h * DH_ + d] += acc;
}

__global__ void resid_add(float* __restrict__ X1, const float* __restrict__ PROJ) {
  int r = blockIdx.x, b = blockIdx.y, t = threadIdx.x;
  size_t xo = ((size_t)b * SEQ1 + r) * C_;
  size_t po = ((size_t)b * NPD + PAD + r) * C_;
  X1[xo + t]       += PROJ[po + t];
  X1[xo + t + 256] += PROJ[po + t + 256];
}

__global__ void ppeg_kernel(const float* __restrict__ X1,
                            const float* __restrict__ w3, const float* __restrict__ b3,
                            const float* __restrict__ w5, const float* __restrict__ b5,
                            const float* __restrict__ w7, const float* __restrict__ b7,
                            float* __restrict__ TMP) {
  int pix = blockIdx.x, b = blockIdx.y, c = threadIdx.x;  // 512 threads
  int i = pix / 100, j = pix % 100;
  const float* F = X1 + ((size_t)b * SEQ1 + 1) * C_;
  auto at = [&](int ii, int jj) -> float {
    if (ii < 0 || ii >= 100 || jj < 0 || jj >= 100) return 0.f;
    return F[((size_t)(ii * 100 + jj)) * C_ + c];
  };
  float a3 = 0.f, a5 = 0.f, a7 = 0.f;
  for (int di = -1; di <= 1; ++di)
    for (int dj = -1; dj <= 1; ++dj)
      a3 += w3[c * 9 + (di + 1) * 3 + (dj + 1)] * at(i + di, j + dj);
  for (int di = -2; di <= 2; ++di)
    for (int dj = -2; dj <= 2; ++dj)
      a5 += w5[c * 25 + (di + 2) * 5 + (dj + 2)] * at(i + di, j + dj);
  for (int di = -3; di <= 3; ++di)
    for (int dj = -3; dj <= 3; ++dj)
      a7 += w7[c * 49 + (di + 3) * 7 + (dj + 3)] * at(i + di, j + dj);
  TMP[((size_t)b * NTOK + pix) * C_ + c] =
      at(i, j) + a3 + b3[c] + a5 + b5[c] + a7 + b7[c];
}

__global__ void ppeg_copyback(float* __restrict__ X1, const float* __restrict__ TMP) {
  int pix = blockIdx.x, b = blockIdx.y, c = threadIdx.x;
  X1[((size_t)b * SEQ1 + 1 + pix) * C_ + c] = TMP[((size_t)b * NTOK + pix) * C_ + c];
}

__global__ void final_head(const float* __restrict__ X1, const float* __restrict__ g,
                           const float* __restrict__ bb, const float* __restrict__ w2,
                           const float* __restrict__ b2, float* __restrict__ out) {
  int b = blockIdx.x, t = threadIdx.x;
  const float* x = X1 + (size_t)b * SEQ1 * C_;  // cls row
  float x0 = x[t], x1 = x[t + 256];
  __shared__ float red[256], r2[256];
  red[t] = x0 + x1; __syncthreads();
  for (int s = 128; s > 0; s >>= 1) { if (t < s) red[t] += red[t + s]; __syncthreads(); }
  float mu = red[0] / 512.f; __syncthreads();
  red[t] = (x0 - mu) * (x0 - mu) + (x1 - mu) * (x1 - mu); __syncthreads();
  for (int s = 128; s > 0; s >>= 1) { if (t < s) red[t] += red[t + s]; __syncthreads(); }
  float inv = rsqrtf(red[0] / 512.f + 1e-5f); __syncthreads();
  float c0 = (x0 - mu) * inv * g[t] + bb[t];
  float c1 = (x1 - mu) * inv * g[t + 256] + bb[t + 256];
  red[t] = c0 * w2[t * 2 + 0] + c1 * w2[(t + 256) * 2 + 0];
  r2[t]  = c0 * w2[t * 2 + 1] + c1 * w2[(t + 256) * 2 + 1];
  __syncthreads();
  for (int s = 128; s > 0; s >>= 1) {
    if (t < s) { red[t] += red[t + s]; r2[t] += r2[t + s]; }
    __syncthreads();
  }
  if (t == 0) {
    float l0 = red[0] + b2[0], l1 = r2[0] + b2[1];
    float mx = fmaxf(l0, l1);
    float e0 = expf(l0 - mx), e1 = expf(l1 - mx), s = e0 + e1;
    out[b * 2 + 0] = l0;  out[b * 2 + 1] = l1;
    out[4 + b * 2 + 0] = e0 / s;  out[4 + b * 2 + 1] = e1 / s;
    out[8 + b] = (l0 >= l1) ? 0.f : 1.f;  // argmax (first index on tie)
  }
}

// ---------------------------------------------------------------------------
// one self-Nystrom layer (X1 += attention(LN(X1)))
// ---------------------------------------------------------------------------
static void run_layer(const float* lng, const float* lnb, const float* qkvw,
                      const float* outw, const float* outb, const float* resw,
                      float* ws, hipStream_t stream) {
  float* X1 = ws + OFF_X1;  float* XP  = ws + OFF_XP;  float* QKV = ws + OFF_QKV;
  float* SIM = ws + OFF_SIM; float* LM = ws + OFF_LM;
  float* A2 = ws + OFF_A2;  float* Z   = ws + OFF_Z;   float* Z2  = ws + OFF_Z2;
  float* G1 = ws + OFF_G1;  float* P   = ws + OFF_P;   float* E   = ws + OFF_E;
  float* T  = ws + OFF_T;   float* U   = ws + OFF_U;   float* SC  = ws + OFF_SC;

  const long sMM = MLM * MLM;               // 65536
  const long sLM = 2 * MLM * DH_;           // 32768 per bh
  const long sT  = MLM * DH_;               // 16384 per bh

  // LN + front-pad -> XP (2 x 10240 x 512)
  ln_pad_kernel<<<dim3(NPD, 2), 256, 0, stream>>>(X1, lng, lnb, XP);

  // QKV = XP @ qkv_w  (20480 x 512 x 1536)
  gemm_nn<<<dim3(96, 160, 1), 256, SMEM_NN, stream>>>(XP, C_, 0, 0, qkvw, 3 * C_, 0, 0,
        QKV, 3 * C_, 0, 0, 20480, 1536, 512, nullptr, 1.f, 0, 1);

  // landmarks (q pre-scaled)
  landmark_kernel<<<dim3(MLM, 16), 64, 0, stream>>>(QKV, LM);

  // sim2 = q_l @ k_l^T (256x256 per bh) -> softmax -> A2
  gemm_nt<<<dim3(16, 2, 16), 256, SMEM_NT, stream>>>(LM, DH_, sLM, 0,
        LM + sT, DH_, sLM, 0, A2, MLM, sMM, 0, 256, 256, 64, 1.f, 1);
  softmax_rows<<<dim3(256, 16), 256, 0, stream>>>(A2, 256, MLM, sMM);

  // Moore-Penrose pinv init
  init_scal<<<1, 8, 0, stream>>>(SC);
  pinv_absmax<<<dim3(256, 16), 256, 0, stream>>>(A2, SC);
  pinv_make_z<<<dim3(256, 16), 256, 0, stream>>>(A2, SC, Z);

  float* zc = Z; float* zn = Z2;
  for (int it = 0; it < 6; ++it) {
    gemm_nn<<<dim3(16, 2, 16), 256, SMEM_NN, stream>>>(A2, MLM, sMM, 0, zc, MLM, sMM, 0,
          G1, MLM, sMM, 0, 256, 256, 256, nullptr, 1.f, 0, 1);
    diag_sub<<<dim3(256, 16), 256, 0, stream>>>(G1, E, 7.f);
    gemm_nn<<<dim3(16, 2, 16), 256, SMEM_NN, stream>>>(G1, MLM, sMM, 0, E, MLM, sMM, 0,
          P, MLM, sMM, 0, 256, 256, 256, nullptr, 1.f, 0, 1);
    diag_sub<<<dim3(256, 16), 256, 0, stream>>>(P, E, 15.f);
    gemm_nn<<<dim3(16, 2, 16), 256, SMEM_NN, stream>>>(G1, MLM, sMM, 0, E, MLM, sMM, 0,
          P, MLM, sMM, 0, 256, 256, 256, nullptr, 1.f, 0, 1);
    diag_sub<<<dim3(256, 16), 256, 0, stream>>>(P, E, 13.f);
    gemm_nn<<<dim3(16, 2, 16), 256, SMEM_NN, stream>>>(zc, MLM, sMM, 0, E, MLM, sMM, 0,
          zn, MLM, sMM, 0, 256, 256, 256, nullptr, 0.25f, 0, 1);
    float* tmp = zc; zc = zn; zn = tmp;
  }
  // zc == a2_inv (6 swaps -> back in Z)

  // sim3 = q_l @ k^T (256 x 10240 per bh) -> softmax
  gemm_nt<<<dim3(640, 2, 16), 256, SMEM_NT, stream>>>(LM, DH_, 8 * sLM, sLM,
        QKV + C_, 3 * C_, NPD * 3 * C_, DH_,
        SIM, NPD, 8 * MLM * NPD, MLM * NPD, 256, 10240, 64, 1.f, 8);
  softmax_rows<<<dim3(256, 16), 256, 0, stream>>>(SIM, 10240, NPD, MLM * NPD);

  // T = a3 @ v (256 x 64, K = 10240)
  gemm_nn<<<dim3(4, 2, 16), 256, SMEM_NN, stream>>>(SIM, NPD, 8 * MLM * NPD, MLM * NPD,
        QKV + 2 * C_, 3 * C_, NPD * 3 * C_, DH_,
        T, DH_, 8 * sT, sT, 256, 64, 10240, nullptr, 1.f, 0, 8);

  // U = a2_inv @ T
  gemm_nn<<<dim3(4, 2, 16), 256, SMEM_NN, stream>>>(zc, MLM, sMM, 0, T, DH_, sT, 0,
        U, DH_, sT, 0, 256, 64, 256, nullptr, 1.f, 0, 1);

  // sim1 = (q * scale) @ k_l^T (10240 x 256 per bh) -> softmax
  gemm_nt<<<dim3(16, 80, 16), 256, SMEM_NT, stream>>>(QKV, 3 * C_, NPD * 3 * C_, DH_,
        LM + sT, DH_, 8 * sLM, sLM,
        SIM, MLM, 8 * NPD * MLM, NPD * MLM, 10240, 256, 64, 0.125f, 8);
  softmax_rows<<<dim3(10240, 16), 256, 0, stream>>>(SIM, 256, MLM, NPD * MLM);

  // out = a1 @ U -> XP region as (b, p, 512), head h at col h*64
  gemm_nn<<<dim3(4, 80, 16), 256, SMEM_NN, stream>>>(SIM, MLM, 8 * NPD * MLM, NPD * MLM,
        U, DH_, 8 * sT, sT,
        XP, C_, NPD * C_, DH_, 10240, 64, 256, nullptr, 1.f, 0, 8);

  // depthwise 33-tap residual conv on v
  res_conv_add<<<dim3(NPD, 16), 64, 0, stream>>>(QKV, resw, XP);

  // projection (20480 x 512 x 512) -> SIM region, then residual add of last 10001 rows
  gemm_nn<<<dim3(32, 160, 1), 256, SMEM_NN, stream>>>(XP, C_, 0, 0, outw, C_, 0, 0,
        SIM, C_, 0, 0, 20480, 512, 512, outb, 1.f, 0, 1);
  resid_add<<<dim3(SEQ1, 2), 256, 0, stream>>>(X1, SIM);
}

// ---------------------------------------------------------------------------
extern "C" void kernel_launch(void* const* d_in, const int* in_sizes, int n_in,
                              void* d_out, int out_size, void* d_ws, size_t ws_size,
                              hipStream_t stream) {
  const float* x      = (const float*)d_in[0];
  const float* fc1w   = (const float*)d_in[1];
  const float* fc1b   = (const float*)d_in[2];
  const float* cls    = (const float*)d_in[3];
  const float* a1lng  = (const float*)d_in[4];
  const float* a1lnb  = (const float*)d_in[5];
  const float* a1qkv  = (const float*)d_in[6];
  const float* a1ow   = (const float*)d_in[7];
  const float* a1ob   = (const float*)d_in[8];
  const float* a1res  = (const float*)d_in[9];
  const float* a2lng  = (const float*)d_in[10];
  const float* a2lnb  = (const float*)d_in[11];
  const float* a2qkv  = (const float*)d_in[12];
  const float* a2ow   = (const float*)d_in[13];
  const float* a2ob   = (const float*)d_in[14];
  const float* a2res  = (const float*)d_in[15];
  const float* c1w    = (const float*)d_in[16];
  const float* c1b    = (const float*)d_in[17];
  const float* c2w    = (const float*)d_in[18];
  const float* c2b    = (const float*)d_in[19];
  const float* c3w    = (const float*)d_in[20];
  const float* c3b    = (const float*)d_in[21];
  const float* lng    = (const float*)d_in[22];
  const float* lnb    = (const float*)d_in[23];
  const float* fc2w   = (const float*)d_in[24];
  const float* fc2b   = (const float*)d_in[25];
  (void)in_sizes; (void)n_in; (void)out_size; (void)ws_size;

  float* ws = (float*)d_ws;

  // fc1 + ReLU, scattered into X1 rows [1..10000] per batch (cls gap at row 0)
  gemm_nn<<<dim3(32, 157, 1), 256, SMEM_NN, stream>>>(x, DIN, 0, 0, fc1w, C_, 0, 0,
        ws + OFF_X1, C_, 0, 0, 20000, 512, 768, fc1b, 1.f, /*relu|remap*/ 3, 1);
  cls_init<<<2, 256, 0, stream>>>(ws + OFF_X1, cls);

  // attention layer 1
  run_layer(a1lng, a1lnb, a1qkv, a1ow, a1ob, a1res, ws, stream);

  // PPEG: feat += dw3 + dw5 + dw7 (each with bias), cls untouched
  ppeg_kernel<<<dim3(NTOK, 2), 512, 0, stream>>>(ws + OFF_X1, c1w, c1b, c2w, c2b,
                                                 c3w, c3b, ws + OFF_SIM);
  ppeg_copyback<<<dim3(NTOK, 2), 512, 0, stream>>>(ws + OFF_X1, ws + OFF_SIM);

  // attention layer 2
  run_layer(a2lng, a2lnb, a2qkv, a2ow, a2ob, a2res, ws, stream);

  // final LN(cls) @ fc2 -> logits, softmax, argmax
  final_head<<<2, 256, 0, stream>>>(ws + OFF_X1, lng, lnb, fc2w, fc2b, (float*)d_out);
}